// LstmDrop_33818572489315
// MI455X (gfx1250) — compile-verified
//
#include <hip/hip_runtime.h>
#include <hip/hip_bf16.h>
#include <math.h>

#define T_STEPS 200
#define B 32
#define H 1024
#define G4H 4096                       // 4*H
#define KG_HALF 256                    // H/4 k-groups per phase (x-phase, h-phase)
#define PW_PER_L (2 * KG_HALF * G4H * 4)  // 8,388,608 floats per layer

typedef __attribute__((ext_vector_type(2))) float v2f;
typedef __attribute__((ext_vector_type(8))) float v8f;

__device__ __forceinline__ float sigmf(float x) { return 1.0f / (1.0f + __expf(-x)); }

// Pack [W_ih ; W_hh] (each [4H,H] row-major) into WMMA-B-native layout:
// PW[kg][n][kk] = W[n][4*kg+kk], kg in [0,512), n in [0,4096), kk in [0,4).
// A lane then reads its (K,K+1) pair for column n as one aligned float2.
__global__ void pack_weights(const float* __restrict__ Wih,
                             const float* __restrict__ Whh,
                             float* __restrict__ PW) {
  unsigned idx = blockIdx.x * blockDim.x + threadIdx.x;  // 16,777,216 total
  int kk = idx & 3;
  int n  = (idx >> 2) & 4095;
  int kg = (idx >> 14) & 511;
  int l  = idx >> 23;
  int k  = kg * 4 + kk;
  float v;
  if (k < H) v = Wih[((size_t)l * G4H + n) * H + k];
  else       v = Whh[((size_t)l * G4H + n) * H + (k - H)];
  PW[idx] = v;
}

// bsum = b_ih + b_hh; copy h0 -> hbuf[l][buf0], c0 -> cbuf.
__global__ void init_state(const float* __restrict__ bih, const float* __restrict__ bhh,
                           const float* __restrict__ h0, const float* __restrict__ c0,
                           float* __restrict__ bsum, float* __restrict__ hbuf,
                           float* __restrict__ cbuf) {
  int idx = blockIdx.x * blockDim.x + threadIdx.x;
  if (idx < 2 * G4H) { bsum[idx] = bih[idx] + bhh[idx]; return; }
  idx -= 2 * G4H;
  if (idx < 2 * B * H) {
    int l = idx / (B * H);
    int off = idx - l * (B * H);
    hbuf[(size_t)(l * 2) * (B * H) + off] = h0[idx];   // buffer 0 of layer l
    return;
  }
  idx -= 2 * B * H;
  if (idx < 2 * B * H) cbuf[idx] = c0[idx];
}

// One fused LSTM step for one layer:
//   gates = [x_t | h_prev] (B x 2048) * PW (2048 x 4096) + bsum
//   c = sig(f)*c + sig(i)*tanh(g);  h = sig(o)*tanh(c);  y = h*mask
// Grid = 128 waves: wave&1 -> M tile (rows 0-15 / 16-31), wave>>1 -> nh tile (64 tiles).
// Each wave accumulates 4 gate tiles (N = nh0 + g*H) sharing the A operand,
// so the pointwise stage has i,f,g,o co-resident per lane slot.
__global__ __launch_bounds__(256) void lstm_step(
    const float* __restrict__ Ax,      // [B,H] layer input at this t
    const float* __restrict__ Hprev,   // [B,H]
    const float* __restrict__ PW,      // packed weights for this layer
    const float* __restrict__ bsum,    // [4H]
    float* __restrict__ Cst,           // [B,H] in/out cell state
    float* __restrict__ Hnext,         // [B,H] out
    const float* __restrict__ omask,   // [B,H] dropout mask slice
    float* __restrict__ Yout,          // [B,H] out (x1 for l=0, y slice for l=1)
    float* __restrict__ HlastOut,      // non-null only at t==T-1
    float* __restrict__ ClastOut) {
  const int tid  = blockIdx.x * blockDim.x + threadIdx.x;
  const int wave = tid >> 5;
  const int lane = threadIdx.x & 31;
  const int half = lane >> 4;        // 0: lanes 0-15, 1: lanes 16-31
  const int lm   = lane & 15;
  const int m0   = (wave & 1) * 16;
  const int nh0  = (wave >> 1) * 16;
  const int row  = m0 + lm;          // A-matrix row this lane supplies
  const int koff = half * 2;         // A/B K offset for this lane half

  v8f acc[4];
#pragma unroll
  for (int g = 0; g < 4; ++g) {
    float bv = bsum[g * H + nh0 + lm];  // bias depends on N only
#pragma unroll
    for (int j = 0; j < 8; ++j) acc[g][j] = bv;
  }

#pragma unroll
  for (int phase = 0; phase < 2; ++phase) {
    const float* A = (phase == 0) ? Ax : Hprev;
    const float* arow  = A + (size_t)row * H + koff;
    const float* bbase = PW + (size_t)phase * (KG_HALF * G4H * 4)
                            + (size_t)(nh0 + lm) * 4 + koff;
#pragma unroll 4
    for (int kgl = 0; kgl < KG_HALF; ++kgl) {
      v2f a = *(const v2f*)(arow + kgl * 4);
#pragma unroll
      for (int g = 0; g < 4; ++g) {
        v2f b = *(const v2f*)(bbase + (size_t)kgl * (G4H * 4) + g * (H * 4));
        acc[g] = __builtin_amdgcn_wmma_f32_16x16x4_f32(
            false, a, false, b, (short)0, acc[g], false, false);
      }
    }
  }

  // Pointwise: C/D tile layout -> lane holds (M = j + 8*half, N = nh0+lm) at slot j.
  const int nh = nh0 + lm;
#pragma unroll
  for (int j = 0; j < 8; ++j) {
    int brow = m0 + j + half * 8;
    int idx  = brow * H + nh;
    float iv = sigmf(acc[0][j]);
    float fv = sigmf(acc[1][j]);
    float gv = tanhf(acc[2][j]);
    float ov = sigmf(acc[3][j]);
    float cn = fv * Cst[idx] + iv * gv;
    float hn = ov * tanhf(cn);
    Cst[idx]   = cn;
    Hnext[idx] = hn;
    Yout[idx]  = hn * omask[idx];
    if (HlastOut) { HlastOut[idx] = hn; ClastOut[idx] = cn; }
  }
}

extern "C" void kernel_launch(void* const* d_in, const int* in_sizes, int n_in,
                              void* d_out, int out_size, void* d_ws, size_t ws_size,
                              hipStream_t stream) {
  (void)in_sizes; (void)n_in; (void)out_size; (void)ws_size;
  const float* x   = (const float*)d_in[0];
  const float* h0  = (const float*)d_in[1];
  const float* c0  = (const float*)d_in[2];
  const float* Wih = (const float*)d_in[3];
  const float* Whh = (const float*)d_in[4];
  const float* bih = (const float*)d_in[5];
  const float* bhh = (const float*)d_in[6];
  const float* mi  = (const float*)d_in[7];  // [1,T,B,H]
  const float* mo  = (const float*)d_in[8];  // [T,B,H]
  float* out = (float*)d_out;

  const size_t BH = (size_t)B * H;
  // Workspace layout (floats): packed weights | bsum | h double-buffers | c | x1
  float* PW   = (float*)d_ws;                 // 2 * 8,388,608
  float* bsum = PW + (size_t)2 * PW_PER_L;    // 2 * 4096
  float* hbuf = bsum + 2 * G4H;               // 2 layers * 2 buffers * BH
  float* cbuf = hbuf + 4 * BH;                // 2 layers * BH
  float* x1   = cbuf + 2 * BH;                // BH

  pack_weights<<<dim3((2 * PW_PER_L) / 256), dim3(256), 0, stream>>>(Wih, Whh, PW);
  init_state<<<dim3((2 * G4H + 4 * (B * H) + 255) / 256), dim3(256), 0, stream>>>(
      bih, bhh, h0, c0, bsum, hbuf, cbuf);

  float* yBase = out;
  float* hOut  = out + (size_t)T_STEPS * BH;   // h_last stack [L,B,H]
  float* cOut  = hOut + 2 * BH;                // c_last stack [L,B,H]

  for (int t = 0; t < T_STEPS; ++t) {
    bool last = (t == T_STEPS - 1);
    // Layer 0: input x[t], output -> x1 (with inter-layer dropout)
    {
      const float* Hp = hbuf + (size_t)(0 * 2 + (t & 1)) * BH;
      float*       Hn = hbuf + (size_t)(0 * 2 + ((t + 1) & 1)) * BH;
      lstm_step<<<dim3(16), dim3(256), 0, stream>>>(
          x + (size_t)t * BH, Hp, PW, bsum, cbuf, Hn,
          mi + (size_t)t * BH, x1,
          last ? hOut : nullptr, last ? cOut : nullptr);
    }
    // Layer 1: input x1, output -> y[t] (with output dropout)
    {
      const float* Hp = hbuf + (size_t)(2 + (t & 1)) * BH;
      float*       Hn = hbuf + (size_t)(2 + ((t + 1) & 1)) * BH;
      lstm_step<<<dim3(16), dim3(256), 0, stream>>>(
          x1, Hp, PW + (size_t)PW_PER_L, bsum + G4H, cbuf + BH, Hn,
          mo + (size_t)t * BH, yBase + (size_t)t * BH,
          last ? hOut + BH : nullptr, last ? cOut + BH : nullptr);
    }
  }
}